// Differentiable_Global_Geometry_PointCloud_15255723836103
// MI455X (gfx1250) — compile-verified
//
#include <hip/hip_runtime.h>

typedef __attribute__((ext_vector_type(2))) float v2f;
typedef __attribute__((ext_vector_type(8))) float v8f;

#define BATCH   2
#define NPTS    4096
#define KNN     20
#define LOCALW  64
#define GRID_PTS (LOCALW * LOCALW)
#define NUM_BFS 32

// CDNA5 async global->LDS path (gate on builtin availability; fall back to
// plain LDS stores if this toolchain doesn't declare them).
#if defined(__has_builtin)
# if __has_builtin(__builtin_amdgcn_global_load_async_to_lds_b64) && \
     __has_builtin(__builtin_amdgcn_s_wait_asynccnt)
#  define USE_ASYNC_LDS 1
# endif
#endif

#ifdef USE_ASYNC_LDS
// b64 payload type as revealed by the builtin's prototype diagnostic:
// pointer to int __attribute__((vector_size(8))) in AS1 (global) / AS3 (LDS).
typedef int a_i2 __attribute__((__vector_size__(2 * sizeof(int))));
typedef __attribute__((address_space(1))) a_i2* a_gp;
typedef __attribute__((address_space(3))) a_i2* a_lp;
#endif

// -------------------------------------------------------------------------
// Kernel 0: zero the output accumulators
// -------------------------------------------------------------------------
__global__ void zero_out_kernel(float* out) {
    if (threadIdx.x < BATCH) out[threadIdx.x] = 0.0f;
}

// -------------------------------------------------------------------------
// Kernel 1: KNN via WMMA f32 16x16x4.
// K slots: K0..K2 = xyz, K3 carries |cand|^2 (A) against -0.5 (B), so the
// WMMA itself produces acc = c.q - |c|^2/2 and the per-query score is just
// -2*acc  (== |c|^2 - 2 c.q, monotone in true distance per query).
// One wave per 16-query tile; per-lane sorted top-20; lane pairs merge in LDS.
// Self distance forced most-negative => self is neighbor 0 (matches ref).
// -------------------------------------------------------------------------
__global__ __launch_bounds__(256) void knn_kernel(const float* __restrict__ pts,
                                                  int* __restrict__ idx_out) {
    __shared__ float sD[8][32][KNN];
    __shared__ int   sI[8][32][KNN];
    const int w    = threadIdx.x >> 5;
    const int lane = threadIdx.x & 31;
    const int col  = lane & 15;
    const int hi   = lane >> 4;
    const int tile = blockIdx.x * 8 + w;                 // 16-query tile id
    const int tiles_per_batch = NPTS / 16;
    const int b  = tile / tiles_per_batch;
    const int qt = tile % tiles_per_batch;
    const int q  = qt * 16 + col;                        // this lane's query
    const float* Pb = pts + (size_t)b * NPTS * 3;

    // B matrix (queries in columns): lanes<16 K=0,1 (x,y); lanes>=16 K=2,3 (z,-0.5)
    const float qx = Pb[q * 3 + 0], qy = Pb[q * 3 + 1], qz = Pb[q * 3 + 2];
    v2f Bm;
    Bm.x = hi ? qz : qx;
    Bm.y = hi ? -0.5f : qy;

    float dl[KNN]; int il[KNN];
#pragma unroll
    for (int j = 0; j < KNN; ++j) { dl[j] = 3.0e38f; il[j] = 0; }

    for (int t = 0; t < NPTS / 16; ++t) {
        const int cand = t * 16 + col;
        __builtin_prefetch(&Pb[(cand + 16) * 3], 0, 0);
        const float cx = Pb[cand * 3 + 0], cy = Pb[cand * 3 + 1], cz = Pb[cand * 3 + 2];
        const float part    = hi ? (cz * cz) : (cx * cx + cy * cy);
        const float sq_full = part + __shfl_xor(part, 16, 32); // |cand|^2, both lanes
        v2f Am;
        Am.x = hi ? cz : cx;
        Am.y = hi ? sq_full : cy;                        // K=3 slot carries |c|^2

        v8f acc = {};
        acc = __builtin_amdgcn_wmma_f32_16x16x4_f32(false, Am, false, Bm,
                                                    (short)0, acc, false, false);
#pragma unroll
        for (int m = 0; m < 8; ++m) {
            const int cm = m + 8 * hi;                   // candidate row in tile
            const int ci = t * 16 + cm;                  // global candidate index
            float d = -2.0f * acc[m];                    // |c|^2 - 2 c.q
            if (ci == q) d = -3.0e38f;                   // self always first (ref diag = -1)
            int id = ci;
            if (d < dl[KNN - 1]) {
#pragma unroll
                for (int j = 0; j < KNN; ++j) {
                    const bool  sw = d < dl[j];
                    const float td = sw ? dl[j] : d;
                    const int   ti = sw ? il[j] : id;
                    dl[j] = sw ? d  : dl[j];
                    il[j] = sw ? id : il[j];
                    d = td; id = ti;
                }
            }
        }
    }
#pragma unroll
    for (int j = 0; j < KNN; ++j) { sD[w][lane][j] = dl[j]; sI[w][lane][j] = il[j]; }
    __syncthreads();
    if (hi == 0) {
        int i0 = 0, i1 = 0;
        int* dst = idx_out + ((size_t)(b * NPTS + q)) * KNN;
#pragma unroll
        for (int j = 0; j < KNN; ++j) {
            float a0 = sD[w][col][i0], a1 = sD[w][col + 16][i1];
            if (a0 <= a1) { dst[j] = sI[w][col][i0]; ++i0; }
            else          { dst[j] = sI[w][col + 16][i1]; ++i1; }
        }
    }
}

// -------------------------------------------------------------------------
// Kernel 2: local covariance + 3x3 Jacobi eigh -> frames (rows = eigvecs,
// ascending eigenvalues; row 0 = normal).
// -------------------------------------------------------------------------
__device__ inline void jacobi3(float A[3][3], float V[3][3]) {
    for (int i = 0; i < 3; ++i)
        for (int j = 0; j < 3; ++j) V[i][j] = (i == j) ? 1.0f : 0.0f;
    for (int sweep = 0; sweep < 8; ++sweep) {
        for (int pq = 0; pq < 3; ++pq) {
            const int p = (pq == 2) ? 1 : 0;
            const int q = (pq == 0) ? 1 : 2;
            float apq = A[p][q];
            if (fabsf(apq) < 1e-20f) continue;
            float tau = (A[q][q] - A[p][p]) / (2.0f * apq);
            float tt  = (tau >= 0.0f ? 1.0f : -1.0f) /
                        (fabsf(tau) + sqrtf(1.0f + tau * tau));
            float c = rsqrtf(1.0f + tt * tt);
            float s = tt * c;
            for (int k = 0; k < 3; ++k) {
                float akp = A[k][p], akq = A[k][q];
                A[k][p] = c * akp - s * akq;
                A[k][q] = s * akp + c * akq;
            }
            for (int k = 0; k < 3; ++k) {
                float apk = A[p][k], aqk = A[q][k];
                A[p][k] = c * apk - s * aqk;
                A[q][k] = s * apk + c * aqk;
            }
            for (int k = 0; k < 3; ++k) {
                float vkp = V[k][p], vkq = V[k][q];
                V[k][p] = c * vkp - s * vkq;
                V[k][q] = s * vkp + c * vkq;
            }
        }
    }
}

__global__ void frames_kernel(const float* __restrict__ pts,
                              const int* __restrict__ idx,
                              float* __restrict__ frames) {
    const int gid = blockIdx.x * blockDim.x + threadIdx.x;
    if (gid >= BATCH * NPTS) return;
    const int b = gid / NPTS;
    const float* Pb = pts + (size_t)b * NPTS * 3;
    const int* nb = idx + (size_t)gid * KNN;
    float px[KNN], py[KNN], pz[KNN];
    float mx = 0.f, my = 0.f, mz = 0.f;
    for (int j = 0; j < KNN; ++j) {
        int m = nb[j];
        px[j] = Pb[m * 3 + 0]; py[j] = Pb[m * 3 + 1]; pz[j] = Pb[m * 3 + 2];
        mx += px[j]; my += py[j]; mz += pz[j];
    }
    mx /= KNN; my /= KNN; mz /= KNN;
    float A[3][3] = {};
    for (int j = 0; j < KNN; ++j) {
        float x = px[j] - mx, y = py[j] - my, z = pz[j] - mz;
        A[0][0] += x * x; A[0][1] += x * y; A[0][2] += x * z;
        A[1][1] += y * y; A[1][2] += y * z; A[2][2] += z * z;
    }
    A[1][0] = A[0][1]; A[2][0] = A[0][2]; A[2][1] = A[1][2];
    for (int i = 0; i < 3; ++i)
        for (int j = 0; j < 3; ++j) A[i][j] *= 0.5f;
    float V[3][3];
    jacobi3(A, V);
    float ev[3] = { A[0][0], A[1][1], A[2][2] };
    int o0 = 0, o1 = 1, o2 = 2;
    if (ev[o0] > ev[o1]) { int t = o0; o0 = o1; o1 = t; }
    if (ev[o1] > ev[o2]) { int t = o1; o1 = o2; o2 = t; }
    if (ev[o0] > ev[o1]) { int t = o0; o0 = o1; o1 = t; }
    const int ord[3] = { o0, o1, o2 };
    float* F = frames + (size_t)gid * 9;
    for (int r = 0; r < 3; ++r)
        for (int c = 0; c < 3; ++c) F[r * 3 + c] = V[c][ord[r]];
}

// -------------------------------------------------------------------------
// Kernel 3: BFS sign disambiguation (frontier/visited/mark shared across
// batch, matching the reference's broadcast semantics).
// -------------------------------------------------------------------------
__global__ void bfs_init(int* visited, int* frontier, int* mark) {
    const int n = blockIdx.x * blockDim.x + threadIdx.x;
    if (n >= NPTS) return;
    visited[n] = 0; mark[n] = 0; frontier[n] = (n == 0) ? 1 : 0;
}

__global__ void bfs_scatter(const int* __restrict__ idx,
                            const int* __restrict__ frontier,
                            float* __restrict__ frames,
                            int* __restrict__ mark) {
    const int gid = blockIdx.x * blockDim.x + threadIdx.x;
    if (gid >= BATCH * NPTS) return;
    const int b = gid / NPTS;
    const int n = gid - b * NPTS;
    if (!frontier[n]) return;
    const int* nb = idx + (size_t)gid * KNN;
    const int s0 = nb[0];                                // == n (self)
    const float* Fr = frames + ((size_t)(b * NPTS + s0)) * 9;
    const float rx = Fr[0], ry = Fr[1], rz = Fr[2];
    for (int j = 0; j < KNN; ++j) {
        int m = nb[j];
        float* Fm = frames + ((size_t)(b * NPTS + m)) * 9;
        float vx = Fm[0], vy = Fm[1], vz = Fm[2];
        float s = (vx * rx + vy * ry + vz * rz) > 0.0f ? 1.0f : -1.0f;
        Fm[0] = s * vx; Fm[1] = s * vy; Fm[2] = s * vz;
        if (j >= 1) mark[m] = 1;
    }
}

__global__ void bfs_update(int* visited, int* frontier, int* mark) {
    const int n = blockIdx.x * blockDim.x + threadIdx.x;
    if (n >= NPTS) return;
    const int v = visited[n] | frontier[n];
    const int f = mark[n] & (v ? 0 : 1);
    visited[n] = v; frontier[n] = f; mark[n] = 0;
}

// -------------------------------------------------------------------------
// Kernel 4: det(frames) and frames[1] *= det
// -------------------------------------------------------------------------
__global__ void det_fix_kernel(float* __restrict__ frames) {
    const int gid = blockIdx.x * blockDim.x + threadIdx.x;
    if (gid >= BATCH * NPTS) return;
    float* F = frames + (size_t)gid * 9;
    float det = F[0] * (F[4] * F[8] - F[5] * F[7])
              - F[1] * (F[3] * F[8] - F[5] * F[6])
              + F[2] * (F[3] * F[7] - F[4] * F[6]);
    F[3] *= det; F[4] *= det; F[5] *= det;
}

// -------------------------------------------------------------------------
// Kernel 5: tangent-plane projection, bounding box, normalized coords
// -------------------------------------------------------------------------
__global__ void prep_kernel(const float* __restrict__ pts,
                            const int* __restrict__ idx,
                            const float* __restrict__ frames,
                            float* __restrict__ dpt_t,
                            float* __restrict__ coord,
                            float* __restrict__ maxlen) {
    const int gid = blockIdx.x * blockDim.x + threadIdx.x;
    if (gid >= BATCH * NPTS) return;
    const int b = gid / NPTS;
    const int n = gid - b * NPTS;
    const float* Pb = pts + (size_t)b * NPTS * 3;
    const float ppx = Pb[n * 3 + 0], ppy = Pb[n * 3 + 1], ppz = Pb[n * 3 + 2];
    const float* F = frames + (size_t)gid * 9;
    const float t1x = F[3], t1y = F[4], t1z = F[5];
    const float t2x = F[6], t2y = F[7], t2z = F[8];
    const int* nb = idx + (size_t)gid * KNN;
    float us[KNN], vs[KNN];
    float umin = 3.0e38f, umax = -3.0e38f, vmin = 3.0e38f, vmax = -3.0e38f;
    for (int j = 0; j < KNN; ++j) {
        int m = nb[j];
        float dx = Pb[m * 3 + 0] - ppx;
        float dy = Pb[m * 3 + 1] - ppy;
        float dz = Pb[m * 3 + 2] - ppz;
        float u = dx * t1x + dy * t1y + dz * t1z;
        float v = dx * t2x + dy * t2y + dz * t2z;
        us[j] = u; vs[j] = v;
        umin = fminf(umin, u); umax = fmaxf(umax, u);
        vmin = fminf(vmin, v); vmax = fmaxf(vmax, v);
    }
    umin *= 1.1f; umax *= 1.1f; vmin *= 1.1f; vmax *= 1.1f;
    const float ml = fmaxf(umax - umin, vmax - vmin);
    maxlen[gid] = ml;
    const float inv = 2.0f / ml;
    for (int j = 0; j < KNN; ++j) {
        dpt_t[((size_t)gid * KNN + j) * 2 + 0] = us[j];
        dpt_t[((size_t)gid * KNN + j) * 2 + 1] = vs[j];
        coord[((size_t)gid * KNN + j) * 2 + 0] = (us[j] - umin) * inv - 1.0f;
        coord[((size_t)gid * KNN + j) * 2 + 1] = (vs[j] - vmin) * inv - 1.0f;
    }
}

// -------------------------------------------------------------------------
// Kernel 6: Voronoi cell counting via WMMA f32 16x16x4.
// Sites staged into LDS with async global->LDS copies (CDNA5 path).
// K3 slot of A carries |c_j|^2 against -0.5 in B, so the WMMA output is
// acc = g.c - |c|^2/2 and the Voronoi score is -2*acc = |c|^2 - 2 g.c.
// Count grid cells where site 0 attains the min (ties -> index 0, i.e. <=).
// -------------------------------------------------------------------------
__global__ __launch_bounds__(256) void voronoi_kernel(const float* __restrict__ coord,
                                                      const float* __restrict__ maxlen,
                                                      float* __restrict__ area) {
    __shared__ float sC[8][KNN][2];
    const int w    = threadIdx.x >> 5;
    const int lane = threadIdx.x & 31;
    const int col  = lane & 15;
    const int hi   = lane >> 4;
    const int gid  = blockIdx.x * 8 + w;                 // point id
    const float* Cg = coord + (size_t)gid * KNN * 2;

#ifdef USE_ASYNC_LDS
    if (lane < KNN) {
        __builtin_amdgcn_global_load_async_to_lds_b64(
            (a_gp)(Cg + lane * 2),
            (a_lp)(&sC[w][lane][0]),
            0, 0);
    }
    __builtin_amdgcn_s_wait_asynccnt(0);
#else
    if (lane < KNN) {
        sC[w][lane][0] = Cg[lane * 2 + 0];
        sC[w][lane][1] = Cg[lane * 2 + 1];
    }
#endif
    __syncthreads();

    // A matrices: lanes<16 carry site (u,v) at K=0,1; lanes>=16 carry
    // (0, |site|^2) at K=2,3.
    v2f A1, A2;
    if (!hi) {
        A1.x = sC[w][col][0]; A1.y = sC[w][col][1];
        if (16 + col < KNN) { A2.x = sC[w][16 + col][0]; A2.y = sC[w][16 + col][1]; }
        else                { A2.x = 0.0f; A2.y = 0.0f; }
    } else {
        const float x = sC[w][col][0], y = sC[w][col][1];
        A1.x = 0.0f; A1.y = x * x + y * y;
        if (16 + col < KNN) {
            const float x2 = sC[w][16 + col][0], y2 = sC[w][16 + col][1];
            A2.x = 0.0f; A2.y = x2 * x2 + y2 * y2;
        } else { A2.x = 0.0f; A2.y = 0.0f; }
    }
    const float c0x = sC[w][0][0], c0y = sC[w][0][1];
    const float sq0 = c0x * c0x + c0y * c0y;
    const float step = 2.0f / (float)(LOCALW - 1);
    int cnt = 0;
    for (int t = 0; t < GRID_PTS / 16; ++t) {
        const int g = t * 16 + col;
        const float gx = -1.0f + step * (float)(g >> 6);
        const float gy = -1.0f + step * (float)(g & 63);
        v2f Bm;
        Bm.x = hi ? 0.0f : gx;
        Bm.y = hi ? -0.5f : gy;
        v8f z = {};
        v8f d1 = __builtin_amdgcn_wmma_f32_16x16x4_f32(false, A1, false, Bm,
                                                       (short)0, z, false, false);
        v8f d2 = __builtin_amdgcn_wmma_f32_16x16x4_f32(false, A2, false, Bm,
                                                       (short)0, z, false, false);
        float mn = 3.0e38f;
#pragma unroll
        for (int m = 0; m < 8; ++m) {
            float s1 = -2.0f * d1[m];
            if (hi == 0 && m == 0) s1 = 3.0e38f;        // exclude site 0 from rest-min
            mn = fminf(mn, s1);
            const bool ok2 = (hi == 0) && (m < KNN - 16); // sites 16..19 valid only
            float s2v = ok2 ? (-2.0f * d2[m]) : 3.0e38f;
            mn = fminf(mn, s2v);
        }
        mn = fminf(mn, __shfl_xor(mn, 16, 32));
        const float s0 = sq0 - 2.0f * (gx * c0x + gy * c0y);
        if (hi == 0 && s0 <= mn) ++cnt;
    }
    cnt += __shfl_xor(cnt, 1, 32);
    cnt += __shfl_xor(cnt, 2, 32);
    cnt += __shfl_xor(cnt, 4, 32);
    cnt += __shfl_xor(cnt, 8, 32);
    if (lane == 0) {
        const float ml = maxlen[gid];
        area[gid] = (float)cnt * ml * ml /
                    (float)((LOCALW - 1) * (LOCALW - 1));
    }
}

// -------------------------------------------------------------------------
// Kernel 7: 2x2 Weingarten fit, Gaussian curvature, Euler accumulation
// -------------------------------------------------------------------------
__global__ void curvature_kernel(const float* __restrict__ frames,
                                 const int* __restrict__ idx,
                                 const float* __restrict__ dpt_t,
                                 const float* __restrict__ area,
                                 float* __restrict__ out) {
    const int gid = blockIdx.x * blockDim.x + threadIdx.x;
    if (gid >= BATCH * NPTS) return;
    const int b = gid / NPTS;
    const float* F = frames + (size_t)gid * 9;
    const float nx = F[0], ny = F[1], nz = F[2];
    const float t1x = F[3], t1y = F[4], t1z = F[5];
    const float t2x = F[6], t2y = F[7], t2z = F[8];
    const int* nb = idx + (size_t)gid * KNN;
    float xx = 0.f, xy = 0.f, yy = 0.f;
    float y00 = 0.f, y01 = 0.f, y10 = 0.f, y11 = 0.f;
    for (int j = 0; j < KNN; ++j) {
        const int m = nb[j];
        const float* Fm = frames + ((size_t)(b * NPTS + m)) * 9;
        const float ax = Fm[0] - nx, ay = Fm[1] - ny, az = Fm[2] - nz;
        const float a = ax * t1x + ay * t1y + az * t1z;
        const float c = ax * t2x + ay * t2y + az * t2z;
        const float u = dpt_t[((size_t)gid * KNN + j) * 2 + 0];
        const float v = dpt_t[((size_t)gid * KNN + j) * 2 + 1];
        xx += u * u; xy += u * v; yy += v * v;
        y00 += a * u; y01 += a * v; y10 += c * u; y11 += c * v;
    }
    xx += 1e-8f; yy += 1e-8f;
    const float det = xx * yy - xy * xy;
    const float inv = 1.0f / det;
    const float w00 = ( y00 * yy - y01 * xy) * inv;
    const float w01 = (-y00 * xy + y01 * xx) * inv;
    const float w10 = ( y10 * yy - y11 * xy) * inv;
    const float w11 = (-y10 * xy + y11 * xx) * inv;
    const float s01 = 0.5f * (w01 + w10);
    const float gauss = w00 * w11 - s01 * s01;
    atomicAdd(&out[b], gauss * area[gid] * 0.15915494309189535f); // 1/(2*pi)
}

// -------------------------------------------------------------------------
// Host launcher
// -------------------------------------------------------------------------
extern "C" void kernel_launch(void* const* d_in, const int* in_sizes, int n_in,
                              void* d_out, int out_size, void* d_ws, size_t ws_size,
                              hipStream_t stream) {
    (void)in_sizes; (void)n_in; (void)out_size; (void)ws_size;
    const float* pts = (const float*)d_in[0];
    float* out = (float*)d_out;

    char* w = (char*)d_ws;
    size_t off = 0;
    auto alloc = [&](size_t bytes) -> void* {
        void* p = (void*)(w + off);
        off = (off + bytes + 255) & ~(size_t)255;
        return p;
    };
    int*   idx      = (int*)  alloc(sizeof(int)   * BATCH * NPTS * KNN);
    float* frames   = (float*)alloc(sizeof(float) * BATCH * NPTS * 9);
    int*   visited  = (int*)  alloc(sizeof(int)   * NPTS);
    int*   frontier = (int*)  alloc(sizeof(int)   * NPTS);
    int*   mark     = (int*)  alloc(sizeof(int)   * NPTS);
    float* dpt      = (float*)alloc(sizeof(float) * BATCH * NPTS * KNN * 2);
    float* coord    = (float*)alloc(sizeof(float) * BATCH * NPTS * KNN * 2);
    float* mlen     = (float*)alloc(sizeof(float) * BATCH * NPTS);
    float* area     = (float*)alloc(sizeof(float) * BATCH * NPTS);

    zero_out_kernel<<<1, 32, 0, stream>>>(out);
    knn_kernel<<<(BATCH * NPTS / 16) / 8, 256, 0, stream>>>(pts, idx);
    frames_kernel<<<(BATCH * NPTS + 255) / 256, 256, 0, stream>>>(pts, idx, frames);
    bfs_init<<<(NPTS + 255) / 256, 256, 0, stream>>>(visited, frontier, mark);
    for (int r = 0; r < NUM_BFS; ++r) {
        bfs_scatter<<<(BATCH * NPTS + 255) / 256, 256, 0, stream>>>(idx, frontier, frames, mark);
        bfs_update<<<(NPTS + 255) / 256, 256, 0, stream>>>(visited, frontier, mark);
    }
    det_fix_kernel<<<(BATCH * NPTS + 255) / 256, 256, 0, stream>>>(frames);
    prep_kernel<<<(BATCH * NPTS + 255) / 256, 256, 0, stream>>>(pts, idx, frames, dpt, coord, mlen);
    voronoi_kernel<<<BATCH * NPTS / 8, 256, 0, stream>>>(coord, mlen, area);
    curvature_kernel<<<(BATCH * NPTS + 255) / 256, 256, 0, stream>>>(frames, idx, dpt, area, out);
}